// Net_46849503265415
// MI455X (gfx1250) — compile-verified
//
#include <hip/hip_runtime.h>

#define FEATURES 4
#define HIDDEN 32

typedef __attribute__((ext_vector_type(2))) float v2f;
typedef __attribute__((ext_vector_type(8))) float v8f;

union V8 { v8f v; float f[8]; };

// ---------------------------------------------------------------- utilities
__global__ void fill_f32(float* __restrict__ p, float val, int n) {
    int i = blockIdx.x * blockDim.x + threadIdx.x;
    if (i < n) p[i] = val;
}

// deg[d] += 1 per edge (deg pre-initialized to 1.0 for the self loop)
__global__ void deg_accum(const long long* __restrict__ dst,
                          float* __restrict__ deg, int E) {
    int e = blockIdx.x * blockDim.x + threadIdx.x;
    if (e < E) atomicAdd(&deg[dst[e]], 1.0f);
}

__global__ void deg_to_dinv(float* __restrict__ dinv, int n) {
    int i = blockIdx.x * blockDim.x + threadIdx.x;
    if (i < n) {
        float d = dinv[i];
        dinv[i] = (d > 0.0f) ? rsqrtf(d) : 0.0f;
    }
}

// ------------------------------------------------ layer-1 GEMM: [N,4]x[4,32]
// One wave = one 16x16 tile.  Writes ht AND seeds acc = ht*dinv^2 + b
// (self-loop + bias fused), so no zero-fill pass is needed before scatter.
__global__ void gemm_k4(const float* __restrict__ x, const float* __restrict__ W,
                        const float* __restrict__ b, const float* __restrict__ dinv,
                        float* __restrict__ ht, float* __restrict__ acc,
                        int nnodes, int row_tiles) {
    int lane  = threadIdx.x & 31;
    int wavei = threadIdx.x >> 5;
    int tile  = blockIdx.x * 8 + wavei;
    if (tile >= row_tiles * 2) return;          // wave-uniform exit, EXEC stays full
    int tile_m = tile >> 1, tile_n = tile & 1;
    int half = lane >> 4, m = lane & 15;
    int col  = tile_n * 16 + m;

    int node  = tile_m * 16 + m;
    int nodec = node < nnodes ? node : nnodes - 1;   // clamp (keeps EXEC all-1s)

    // A 16x4: lanes 0-15 hold K={0,1}, lanes 16-31 hold K={2,3}
    v2f a = *(const v2f*)(x + (size_t)nodec * FEATURES + 2 * half);
    // B 4x16: row K striped across 16 lanes
    v2f bw;
    bw.x = W[(2 * half + 0) * HIDDEN + col];
    bw.y = W[(2 * half + 1) * HIDDEN + col];

    V8 c = {};
    c.v = __builtin_amdgcn_wmma_f32_16x16x4_f32(false, a, false, bw,
                                                (short)0, c.v, false, false);

    float bcol = b[col];
    bool  full = (tile_m * 16 + 16 <= nnodes);       // wave-uniform fast path
    if (full) {
#pragma unroll
        for (int r = 0; r < 8; ++r) {
            int   row = tile_m * 16 + r + 8 * half;
            float di  = dinv[row];
            float v   = c.f[r];
            ht [(size_t)row * HIDDEN + col] = v;
            acc[(size_t)row * HIDDEN + col] = v * di * di + bcol;
        }
    } else {
#pragma unroll
        for (int r = 0; r < 8; ++r) {
            int row = tile_m * 16 + r + 8 * half;
            if (row < nnodes) {
                float di = dinv[row];
                float v  = c.f[r];
                ht [(size_t)row * HIDDEN + col] = v;
                acc[(size_t)row * HIDDEN + col] = v * di * di + bcol;
            }
        }
    }
}

// ------------------------------------------- layer-2/3 GEMM: [N,32]x[32,32]
// W (4KB) staged in LDS once per block; 8 chained K=4 WMMAs per tile.
__global__ void gemm_k32(const float* __restrict__ A, const float* __restrict__ W,
                         const float* __restrict__ b, const float* __restrict__ dinv,
                         float* __restrict__ ht, float* __restrict__ acc,
                         int nnodes, int row_tiles) {
    __shared__ __attribute__((aligned(16))) float sW[HIDDEN * HIDDEN];
    {
        const float4* Wv  = (const float4*)W;
        float4*       sWv = (float4*)sW;
        sWv[threadIdx.x] = Wv[threadIdx.x];      // 256 threads x float4 = 1024 f32
    }
    __syncthreads();

    int lane  = threadIdx.x & 31;
    int wavei = threadIdx.x >> 5;
    int tile  = blockIdx.x * 8 + wavei;
    if (tile >= row_tiles * 2) return;
    int tile_m = tile >> 1, tile_n = tile & 1;
    int half = lane >> 4, m = lane & 15;
    int col  = tile_n * 16 + m;

    int node  = tile_m * 16 + m;
    int nodec = node < nnodes ? node : nnodes - 1;
    const float* arow = A + (size_t)nodec * HIDDEN;

    V8 c = {};
#pragma unroll
    for (int s = 0; s < 8; ++s) {
        int k0 = 4 * s + 2 * half;
        v2f a = *(const v2f*)(arow + k0);        // 8B-aligned
        v2f bw;
        bw.x = sW[(k0 + 0) * HIDDEN + col];      // conflict-free ds reads
        bw.y = sW[(k0 + 1) * HIDDEN + col];
        c.v = __builtin_amdgcn_wmma_f32_16x16x4_f32(false, a, false, bw,
                                                    (short)0, c.v, false, false);
    }

    float bcol = b[col];
    bool  full = (tile_m * 16 + 16 <= nnodes);
    if (full) {
#pragma unroll
        for (int r = 0; r < 8; ++r) {
            int   row = tile_m * 16 + r + 8 * half;
            float di  = dinv[row];
            float v   = c.f[r];
            ht [(size_t)row * HIDDEN + col] = v;
            acc[(size_t)row * HIDDEN + col] = v * di * di + bcol;
        }
    } else {
#pragma unroll
        for (int r = 0; r < 8; ++r) {
            int row = tile_m * 16 + r + 8 * half;
            if (row < nnodes) {
                float di = dinv[row];
                float v  = c.f[r];
                ht [(size_t)row * HIDDEN + col] = v;
                acc[(size_t)row * HIDDEN + col] = v * di * di + bcol;
            }
        }
    }
}

// ----------------------------- scatter: acc[dst] += ht[src] * dinv[s]*dinv[d]
// One thread per edge: indices read once, row gathered as 8 x float4,
// scattered via global_atomic_add_f32.
__global__ void scatter32(const float* __restrict__ ht,
                          const long long* __restrict__ src,
                          const long long* __restrict__ dst,
                          const float* __restrict__ dinv,
                          float* __restrict__ acc, int E) {
    int e = blockIdx.x * blockDim.x + threadIdx.x;
    if (e >= E) return;
    long long s = src[e], d = dst[e];
    float w = dinv[s] * dinv[d];
    const float* hs = ht + (size_t)s * HIDDEN;
    float*       o  = acc + (size_t)d * HIDDEN;
#pragma unroll
    for (int ch = 0; ch < HIDDEN; ch += 4) {
        float4 v = *(const float4*)(hs + ch);
        atomicAdd(o + ch + 0, v.x * w);
        atomicAdd(o + ch + 1, v.y * w);
        atomicAdd(o + ch + 2, v.z * w);
        atomicAdd(o + ch + 3, v.w * w);
    }
}

// vectorized ReLU: h = max(acc, 0), float4 per thread
__global__ void relu32(const float* __restrict__ acc, float* __restrict__ hout,
                       int nvec4) {
    int i = blockIdx.x * blockDim.x + threadIdx.x;
    if (i >= nvec4) return;
    float4 v = ((const float4*)acc)[i];
    v.x = fmaxf(v.x, 0.0f);
    v.y = fmaxf(v.y, 0.0f);
    v.z = fmaxf(v.z, 0.0f);
    v.w = fmaxf(v.w, 0.0f);
    ((float4*)hout)[i] = v;
}

// ------------------------------------------------------- layer 4 (out dim 1)
// t[n] = h[n].W3 ; out[n] seeded with self-loop + bias (scatter adds on top)
__global__ void dot32(const float* __restrict__ h, const float* __restrict__ W3,
                      const float* __restrict__ b3, const float* __restrict__ dinv,
                      float* __restrict__ t, float* __restrict__ out, int nnodes) {
    int n = blockIdx.x * blockDim.x + threadIdx.x;
    if (n >= nnodes) return;
    const float* r = h + (size_t)n * HIDDEN;
    float acc = 0.0f;
#pragma unroll
    for (int c = 0; c < HIDDEN; c += 4) {
        float4 a = *(const float4*)(r + c);
        float4 w = *(const float4*)(W3 + c);
        acc += a.x * w.x + a.y * w.y + a.z * w.z + a.w * w.w;
    }
    float di = dinv[n];
    t[n]   = acc;
    out[n] = acc * di * di + b3[0];
}

__global__ void scatter1(const float* __restrict__ t,
                         const long long* __restrict__ src,
                         const long long* __restrict__ dst,
                         const float* __restrict__ dinv,
                         float* __restrict__ out, int E) {
    int e = blockIdx.x * blockDim.x + threadIdx.x;
    if (e >= E) return;
    long long s = src[e], d = dst[e];
    atomicAdd(&out[d], t[s] * dinv[s] * dinv[d]);
}

// ---------------------------------------------------------------- launcher
extern "C" void kernel_launch(void* const* d_in, const int* in_sizes, int n_in,
                              void* d_out, int out_size, void* d_ws, size_t ws_size,
                              hipStream_t stream) {
    const float*     x   = (const float*)d_in[0];
    const long long* ei  = (const long long*)d_in[1];   // int64 edge_index [2,E]
    const float*     W1  = (const float*)d_in[2];
    const float*     b1  = (const float*)d_in[3];
    const float*     W2  = (const float*)d_in[4];
    const float*     b2  = (const float*)d_in[5];
    const float*     W21 = (const float*)d_in[6];
    const float*     b21 = (const float*)d_in[7];
    const float*     W3  = (const float*)d_in[8];
    const float*     b3  = (const float*)d_in[9];
    float*           out = (float*)d_out;

    const int N = in_sizes[0] / FEATURES;
    const int E = in_sizes[1] / 2;
    const long long* src = ei;
    const long long* dst = ei + E;

    float* ws   = (float*)d_ws;
    float* dinv = ws;                 ws += N;                       // [N]
    float* t    = ws;                 ws += N;                       // [N]
    float* h    = ws;                 ws += (size_t)N * HIDDEN;      // [N,32]
    float* ht   = ws;                 ws += (size_t)N * HIDDEN;      // [N,32]
    float* acc  = ws;                                                // [N,32]

    const int B = 256;
    auto g = [](int n) { return (n + 255) / 256; };

    const int row_tiles   = (N + 15) / 16;
    const int gemm_blocks = (row_tiles * 2 + 7) / 8;     // 8 waves / block
    const int nvec4       = N * HIDDEN / 4;

    // ---- symmetric normalization: dinv = rsqrt(1 + in-degree)
    fill_f32<<<g(N), B, 0, stream>>>(dinv, 1.0f, N);
    deg_accum<<<g(E), B, 0, stream>>>(dst, dinv, E);
    deg_to_dinv<<<g(N), B, 0, stream>>>(dinv, N);

    // ---- layer 1: 4 -> 32, ReLU
    gemm_k4<<<gemm_blocks, B, 0, stream>>>(x, W1, b1, dinv, ht, acc, N, row_tiles);
    scatter32<<<g(E), B, 0, stream>>>(ht, src, dst, dinv, acc, E);
    relu32<<<g(nvec4), B, 0, stream>>>(acc, h, nvec4);

    // ---- layer 2: 32 -> 32, ReLU
    gemm_k32<<<gemm_blocks, B, 0, stream>>>(h, W2, b2, dinv, ht, acc, N, row_tiles);
    scatter32<<<g(E), B, 0, stream>>>(ht, src, dst, dinv, acc, E);
    relu32<<<g(nvec4), B, 0, stream>>>(acc, h, nvec4);

    // ---- layer 3: 32 -> 32, ReLU
    gemm_k32<<<gemm_blocks, B, 0, stream>>>(h, W21, b21, dinv, ht, acc, N, row_tiles);
    scatter32<<<g(E), B, 0, stream>>>(ht, src, dst, dinv, acc, E);
    relu32<<<g(nvec4), B, 0, stream>>>(acc, h, nvec4);

    // ---- layer 4: 32 -> 1 (no ReLU); out seeded by dot32, edges add on top
    dot32<<<g(N), B, 0, stream>>>(h, W3, b3, dinv, t, out, N);
    scatter1<<<g(E), B, 0, stream>>>(t, src, dst, dinv, out, E);
}